// TestBatchedSoftNMS_3040836846184
// MI455X (gfx1250) — compile-verified
//
#include <hip/hip_runtime.h>
#include <hip/hip_bf16.h>

// Batched Gaussian soft-NMS (sigma=0.5, thr=0.05) for MI455X / gfx1250.
// Design: inherently sequential scan (3000 steps) -> single workgroup,
// all state in WGP LDS (~84KB of 320KB), wave32 shuffle argmax, 3 barriers
// per step. Initial bulk load uses CDNA5 async global->LDS ops (ASYNCcnt).

#define NBOX   3008          // capacity (>= 3000), keeps arrays aligned
#define BS     512           // 16 wave32 waves on one WGP
#define NWAVE  (BS / 32)

struct __align__(16) F4 { float x, y, z, w; };

// Types matching the clang builtin signatures discovered in round 1:
//   __builtin_amdgcn_global_load_async_to_lds_b128(v4i AS1*, v4i AS3*, Imm, Imm)
//   __builtin_amdgcn_global_load_async_to_lds_b32 (int AS1*, int AS3*, Imm, Imm)
typedef int v4i __attribute__((vector_size(16)));
typedef __attribute__((address_space(1))) v4i* g_v4i_p;
typedef __attribute__((address_space(3))) v4i* l_v4i_p;
typedef __attribute__((address_space(1))) int* g_i_p;
typedef __attribute__((address_space(3))) int* l_i_p;

__global__ __launch_bounds__(BS, 1)
void softnms_gfx1250_kernel(const float* __restrict__ boxes,
                            const float* __restrict__ scores,
                            const int*   __restrict__ idxs,
                            float*       __restrict__ out,
                            int n)
{
    __shared__ F4    sbox[NBOX];    // offset boxes (xyxy)
    __shared__ float sarea[NBOX];   // per-box area of offset boxes
    __shared__ float ssc[NBOX];     // scores (decayed in place)
    __shared__ int   sord[NBOX];    // order permutation
    __shared__ float redv[NWAVE];
    __shared__ int   redi[NWAVE];
    __shared__ F4    selbox;
    __shared__ float selarea;
    __shared__ float smaxc;

    const int tid  = (int)threadIdx.x;
    const int lane = tid & 31;
    const int wid  = tid >> 5;

    // ---------------- Phase 0: bulk load global -> LDS ----------------
#if defined(__gfx1250__) && \
    __has_builtin(__builtin_amdgcn_global_load_async_to_lds_b128) && \
    __has_builtin(__builtin_amdgcn_global_load_async_to_lds_b32)
    for (int k = tid; k < n; k += BS) {
        __builtin_amdgcn_global_load_async_to_lds_b128(
            (g_v4i_p)(boxes + 4 * k), (l_v4i_p)(&sbox[k]), 0, 0);
        __builtin_amdgcn_global_load_async_to_lds_b32(
            (g_i_p)(scores + k), (l_i_p)(&ssc[k]), 0, 0);
    }
#if __has_builtin(__builtin_amdgcn_s_wait_asynccnt)
    __builtin_amdgcn_s_wait_asynccnt(0);
#else
    asm volatile("s_wait_asynccnt 0" ::: "memory");
#endif
#else
    for (int k = tid; k < n; k += BS) {
        F4 b;
        b.x = boxes[4 * k + 0];
        b.y = boxes[4 * k + 1];
        b.z = boxes[4 * k + 2];
        b.w = boxes[4 * k + 3];
        sbox[k] = b;
        ssc[k]  = scores[k];
    }
#endif
    __syncthreads();

    // ---------------- Phase 1: max coordinate over all boxes ----------------
    float m = -__builtin_inff();
    for (int k = tid; k < n; k += BS) {
        F4 b = sbox[k];
        m = fmaxf(m, fmaxf(fmaxf(b.x, b.y), fmaxf(b.z, b.w)));
    }
    for (int off = 16; off > 0; off >>= 1)
        m = fmaxf(m, __shfl_down(m, off, 32));
    if (lane == 0) redv[wid] = m;
    __syncthreads();
    if (tid == 0) {
        float mm = redv[0];
        for (int w = 1; w < NWAVE; ++w) mm = fmaxf(mm, redv[w]);
        smaxc = mm;
    }
    __syncthreads();
    const float moff = smaxc + 1.0f;

    // ---------------- Phase 2: per-class offsets, areas, order ----------------
    for (int k = tid; k < n; k += BS) {
        float off = (float)idxs[k] * moff;
        F4 b = sbox[k];
        b.x += off; b.y += off; b.z += off; b.w += off;
        sbox[k]  = b;
        sarea[k] = (b.z - b.x) * (b.w - b.y);
        sord[k]  = k;
    }
    __syncthreads();

    // ---------------- Phase 3: sequential soft-NMS scan ----------------
    for (int i = 0; i < n; ++i) {
        // argmax over slots [i, n): thread scan -> wave32 shuffle reduce
        float bs = -__builtin_inff();
        int   bj = 0x7fffffff;
        for (int k = tid; k < n; k += BS) {
            if (k >= i) {
                float v = ssc[k];
                if (v > bs) { bs = v; bj = k; }   // ascending k => first max kept
            }
        }
        for (int off = 16; off > 0; off >>= 1) {
            float ov = __shfl_down(bs, off, 32);
            int   oj = __shfl_down(bj, off, 32);
            if (ov > bs || (ov == bs && oj < bj)) { bs = ov; bj = oj; }
        }
        if (lane == 0) { redv[wid] = bs; redi[wid] = bj; }
        __syncthreads();

        // thread 0: final reduce across waves, swap slots i<->j, broadcast
        if (tid == 0) {
            float vb = redv[0];
            int   jb = redi[0];
            for (int w = 1; w < NWAVE; ++w) {
                float v = redv[w];
                int   j = redi[w];
                if (v > vb || (v == vb && j < jb)) { vb = v; jb = j; }
            }
            const int j = jb;
            F4 bi = sbox[i], bj4 = sbox[j];
            sbox[i] = bj4; sbox[j] = bi;
            float ai = sarea[i], aj = sarea[j];
            sarea[i] = aj; sarea[j] = ai;
            float si = ssc[i], sj = ssc[j];
            ssc[i] = sj; ssc[j] = si;
            int oi = sord[i], oj = sord[j];
            sord[i] = oj; sord[j] = oi;
            selbox  = bj4;   // box now in slot i
            selarea = aj;
        }
        __syncthreads();

        // gaussian decay of all later slots: s *= exp(-2*iou^2)
        const F4    sb = selbox;
        const float sa = selarea;
        for (int k = tid; k < n; k += BS) {
            if (k > i) {
                F4 c = sbox[k];
                float ix = fminf(sb.z, c.z) - fmaxf(sb.x, c.x);
                float iy = fminf(sb.w, c.w) - fmaxf(sb.y, c.y);
                float inter = fmaxf(ix, 0.0f) * fmaxf(iy, 0.0f);
                float iou = inter / (sa + sarea[k] - inter);
                ssc[k] = ssc[k] * expf(-2.0f * (iou * iou));
            }
        }
        __syncthreads();
    }

    // ---------------- Phase 4: write outputs ----------------
    // d_out = [scores (f32) | order (as f32) | keep_mask (as 0/1 f32)]
    for (int k = tid; k < n; k += BS) {
        float s = ssc[k];
        out[k]         = s;
        out[n + k]     = (float)sord[k];
        out[2 * n + k] = (s > 0.05f) ? 1.0f : 0.0f;
    }
}

extern "C" void kernel_launch(void* const* d_in, const int* in_sizes, int n_in,
                              void* d_out, int out_size, void* d_ws, size_t ws_size,
                              hipStream_t stream) {
    const float* boxes  = (const float*)d_in[0];   // [N,4] f32
    const float* scores = (const float*)d_in[1];   // [N]   f32
    const int*   idxs   = (const int*)d_in[2];     // [N]   i32
    int n = in_sizes[1];
    if (n > 3000) n = 3000;
    softnms_gfx1250_kernel<<<1, BS, 0, stream>>>(boxes, scores, idxs,
                                                 (float*)d_out, n);
}